// MLPClassifier2_62869731279430
// MI455X (gfx1250) — compile-verified
//
#include <hip/hip_runtime.h>
#include <hip/hip_bf16.h>

typedef __attribute__((ext_vector_type(2))) float v2f;
typedef __attribute__((ext_vector_type(8))) float v8f;

#define DIM 1024
#define LEN 32768
#define WIN 5

// ---- workspace layout (float-element offsets into d_ws) ----
#define WS_VWT   0                       // 1024 f : v = W_h^T @ wt_aa
#define WS_VMT   1024                    // 1024 f
#define WS_SCW   2048                    // 32768 f : wt scores
#define WS_SCM   (2048 + 32768)          // 32768 f : mt scores
#define WS_IDXW  (2048 + 2*32768)        // 8 ints
#define WS_IDXM  (WS_IDXW + 8)           // 8 ints
#define WS_FULLW (WS_IDXW + 128)         // 16*1024 f (rows 0-4: h(top), 5: aa, 6-15: 0)
#define WS_FULLM (WS_FULLW + 16*1024)    // 16*1024 f

// ---------------------------------------------------------------------------
// Kernel A: v_wt[j] = sum_i W_h[i][j] * wt_aa[i]   (and same for mt)
// One pass over W_h (4MB), coalesced in j. aa vectors staged in LDS.
// ---------------------------------------------------------------------------
__global__ void k_vproj(const float* __restrict__ wt, const float* __restrict__ mt,
                        const float* __restrict__ Wh, const int* __restrict__ aa_p,
                        float* __restrict__ vwt, float* __restrict__ vmt) {
    __shared__ float awt[DIM];
    __shared__ float amt[DIM];
    const int aa = aa_p[0];
    for (int t = threadIdx.x; t < DIM; t += blockDim.x) {
        awt[t] = wt[(size_t)aa * DIM + t];
        amt[t] = mt[(size_t)aa * DIM + t];
    }
    __syncthreads();
    const int j = blockIdx.x * blockDim.x + threadIdx.x;
    float aw = 0.f, am = 0.f;
    for (int i = 0; i < DIM; ++i) {
        const float w = Wh[(size_t)i * DIM + j];
        aw = fmaf(w, awt[i], aw);
        am = fmaf(w, amt[i], am);
    }
    vwt[j] = aw;
    vmt[j] = am;
}

// ---------------------------------------------------------------------------
// Kernel B: scores[l] = emb[l] . v        (bandwidth-critical: 256MB stream)
// One wave (32 lanes) per 4KB row; float4 loads; v in LDS; shuffle-reduce.
// ---------------------------------------------------------------------------
__global__ void k_scores(const float* __restrict__ wt, const float* __restrict__ mt,
                         const float* __restrict__ vwt, const float* __restrict__ vmt,
                         float* __restrict__ scw, float* __restrict__ scm) {
    __shared__ float vs[DIM];
    const float* emb = blockIdx.y ? mt : wt;
    const float* v   = blockIdx.y ? vmt : vwt;
    float*       sc  = blockIdx.y ? scm : scw;
    for (int t = threadIdx.x; t < DIM; t += blockDim.x) vs[t] = v[t];
    __syncthreads();

    const int wave = threadIdx.x >> 5;
    const int lane = threadIdx.x & 31;
    const int row  = blockIdx.x * 8 + wave;

    const float4* rp = (const float4*)(emb + (size_t)row * DIM);
    const float4* vp = (const float4*)vs;
    // Hint the next block's row into L2 (speculative; safe if OOB).
    __builtin_prefetch((const void*)(emb + (size_t)(row + 8) * DIM + lane * 4), 0, 0);

    float acc = 0.f;
#pragma unroll
    for (int i = 0; i < 8; ++i) {
        const float4 a = rp[lane + 32 * i];
        const float4 b = vp[lane + 32 * i];
        acc = fmaf(a.x, b.x, acc);
        acc = fmaf(a.y, b.y, acc);
        acc = fmaf(a.z, b.z, acc);
        acc = fmaf(a.w, b.w, acc);
    }
#pragma unroll
    for (int m = 16; m >= 1; m >>= 1) acc += __shfl_xor(acc, m, 32);
    if (lane == 0) sc[row] = acc;
}

// ---------------------------------------------------------------------------
// Kernel C: top-5 argmax (descending, ties -> smaller index, matching
// jax.lax.top_k). 5 full-scan passes over 32768 scores by one 1024-thread
// block per stream; packed (sortable_float_key << 32) | ~idx max-reduction.
// ---------------------------------------------------------------------------
__global__ void k_top5(const float* __restrict__ scw, const float* __restrict__ scm,
                       int* __restrict__ idxw, int* __restrict__ idxm) {
    __shared__ unsigned long long red[1024];
    __shared__ int chosen[WIN];
    const float* sc  = blockIdx.x ? scm : scw;
    int*         out = blockIdx.x ? idxm : idxw;
    const int t = threadIdx.x;

    for (int p = 0; p < WIN; ++p) {
        unsigned long long best = 0ull;
        for (int c = 0; c < LEN / 1024; ++c) {
            const int idx = t + c * 1024;
            bool skip = false;
            for (int q = 0; q < p; ++q) skip |= (chosen[q] == idx);
            if (!skip) {
                const unsigned int u   = __float_as_uint(sc[idx]);
                const unsigned int key = u ^ ((u >> 31) ? 0xFFFFFFFFu : 0x80000000u);
                const unsigned long long pk =
                    ((unsigned long long)key << 32) | (unsigned int)(~idx);
                if (pk > best) best = pk;
            }
        }
        red[t] = best;
        __syncthreads();
        for (int s = 512; s > 0; s >>= 1) {
            if (t < s) {
                const unsigned long long o = red[t + s];
                if (o > red[t]) red[t] = o;
            }
            __syncthreads();
        }
        if (t == 0) {
            const int idx = (int)(~((unsigned int)red[0]));
            chosen[p] = idx;
            out[p]    = idx;
        }
        __syncthreads();
    }
}

// ---------------------------------------------------------------------------
// Kernel D: full[r][i] for r<5: (W_h @ emb[idx_r])[i] + b_h[i]; r==5: aa[i];
// rows 6..15 zero-padded.  Uses V_WMMA_F32_16X16X4_F32:
//   D(16x16) += A(16x4)=W_h[i0:i0+16, k:k+4]  x  B(4x16)=E[k:k+4, 0:16]
// A lane layout (ISA 7.12.2): lanes0-15 hold K=k,k+1; lanes16-31 hold K=k+2,k+3.
// B assumed mirrored: v0/v1 lanes0-15 = rows K=k,k+1 (N=lane),
//                     lanes16-31 = rows K=k+2,k+3 (N=lane-16).
// C/D: VGPR g, lanes0-15 -> (M=g, N=lane); lanes16-31 -> (M=g+8, N=lane-16).
// One wave per block; no divergent control flow around the WMMA (EXEC all-1s).
// ---------------------------------------------------------------------------
__global__ void k_project(const float* __restrict__ wt, const float* __restrict__ mt,
                          const float* __restrict__ Wh, const float* __restrict__ bh,
                          const int* __restrict__ aa_p,
                          const int* __restrict__ idxw, const int* __restrict__ idxm,
                          float* __restrict__ fullw, float* __restrict__ fullm) {
    const float* emb  = blockIdx.y ? mt : wt;
    const int*   idxs = blockIdx.y ? idxm : idxw;
    float*       full = blockIdx.y ? fullm : fullw;

    const int lane = threadIdx.x;       // 32 threads
    const int half = lane >> 4;         // K sub-block select (0: k,k+1 ; 1: k+2,k+3)
    const int l15  = lane & 15;
    const int i0   = blockIdx.x * 16;
    const int aa   = aa_p[0];

    const bool  bvalid = (l15 < WIN);
    const int   myIdx  = bvalid ? idxs[l15] : 0;   // idx 0 is a safe dummy row
    const float* bsrc  = emb + (size_t)myIdx * DIM;
    const float* asrc  = Wh + (size_t)(i0 + l15) * DIM;

    v8f acc = {};
    for (int k = 0; k < DIM; k += 4) {
        const int koff = k + half * 2;
        const float2 av = *(const float2*)(asrc + koff);
        const float2 bv = *(const float2*)(bsrc + koff);
        v2f a; a[0] = av.x; a[1] = av.y;
        v2f b; b[0] = bvalid ? bv.x : 0.f; b[1] = bvalid ? bv.y : 0.f;
        acc = __builtin_amdgcn_wmma_f32_16x16x4_f32(
            /*neg_a=*/false, a, /*neg_b=*/false, b,
            /*c_mod=*/(short)0, acc, /*reuse_a=*/false, /*reuse_b=*/false);
    }

    // D[M][N] = H[i0+M][row=N];  full is row-major [16][DIM]
    const int row = l15;
#pragma unroll
    for (int g = 0; g < 8; ++g) {
        const int col = i0 + half * 8 + g;
        float val;
        if (row < WIN)       val = acc[g] + bh[col];
        else if (row == WIN) val = emb[(size_t)aa * DIM + col];
        else                 val = 0.f;
        full[(size_t)row * DIM + col] = val;
    }
}

// ---------------------------------------------------------------------------
// Kernel E: out(6x6, padded to 16x16) = wt_full @ mt_full^T over K=1024,
// V_WMMA_F32_16X16X4_F32 chain (256 accumulating WMMAs), single wave.
// ---------------------------------------------------------------------------
__global__ void k_final(const float* __restrict__ fullw, const float* __restrict__ fullm,
                        float* __restrict__ out) {
    const int lane = threadIdx.x;
    const int half = lane >> 4;
    const int l15  = lane & 15;

    const float* arow = fullw + (size_t)l15 * DIM;   // A[M=l15][k..]
    const float* brow = fullm + (size_t)l15 * DIM;   // B[K][N=l15] = mt_full[l15][K]

    v8f acc = {};
    for (int k = 0; k < DIM; k += 4) {
        const int koff = k + half * 2;
        const float2 av = *(const float2*)(arow + koff);
        const float2 bv = *(const float2*)(brow + koff);
        v2f a; a[0] = av.x; a[1] = av.y;
        v2f b; b[0] = bv.x; b[1] = bv.y;
        acc = __builtin_amdgcn_wmma_f32_16x16x4_f32(
            false, a, false, b, (short)0, acc, false, false);
    }
    // out[r*6+c] = D[M=r][N=c], r,c in 0..5 -> lanes 0..5 (half 0), VGPRs g=0..5
    if (half == 0 && l15 < 6) {
#pragma unroll
        for (int g = 0; g < 6; ++g) out[g * 6 + l15] = acc[g];
    }
}

// ---------------------------------------------------------------------------
extern "C" void kernel_launch(void* const* d_in, const int* in_sizes, int n_in,
                              void* d_out, int out_size, void* d_ws, size_t ws_size,
                              hipStream_t stream) {
    const float* wt = (const float*)d_in[0];
    const float* mt = (const float*)d_in[1];
    const float* Wh = (const float*)d_in[2];
    const float* bh = (const float*)d_in[3];
    const int*   aa = (const int*)d_in[4];
    // d_in[5] = label, unused.

    float* W    = (float*)d_ws;
    float* vwt  = W + WS_VWT;
    float* vmt  = W + WS_VMT;
    float* scw  = W + WS_SCW;
    float* scm  = W + WS_SCM;
    int* idxw   = (int*)(W + WS_IDXW);
    int* idxm   = (int*)(W + WS_IDXM);
    float* fullw = W + WS_FULLW;
    float* fullm = W + WS_FULLM;
    float* out  = (float*)d_out;

    k_vproj  <<<dim3(DIM / 256),     256,  0, stream>>>(wt, mt, Wh, aa, vwt, vmt);
    k_scores <<<dim3(LEN / 8, 2),    256,  0, stream>>>(wt, mt, vwt, vmt, scw, scm);
    k_top5   <<<dim3(2),             1024, 0, stream>>>(scw, scm, idxw, idxm);
    k_project<<<dim3(DIM / 16, 2),   32,   0, stream>>>(wt, mt, Wh, bh, aa, idxw, idxm,
                                                        fullw, fullm);
    k_final  <<<dim3(1),             32,   0, stream>>>(fullw, fullm, out);
}